// QuanvolutionClassifier_65481071410297
// MI455X (gfx1250) — compile-verified
//
#include <hip/hip_runtime.h>
#include <hip/hip_bf16.h>
#include <math.h>

// CDNA5 / gfx1250 fused quanvolution classifier, round 4.
// Stage 1 (block = 16 images): per patch, WMMA computes  Z = U x S^T  with
// A = U (rows = output basis e) and B = S^T (cols = images); D gives each
// lane its OWN image's amplitudes (lane=image, e = vgpr + 8*half).
// PauliZ measurement = compile-time-signed packed adder tree over the 8
// slots + 4 cross-half swaps done with v_permlanex16_b32 (pure VALU, no
// LDS bpermute / dscnt waits), then ONE float4 LDS store per image.
// Stage 2: logits = feat(16x784) @ W^T (784x10 pad 16), K split over 8
// waves + ds_add_f32 reduction, bias + log_softmax in wave 0.

typedef __attribute__((ext_vector_type(2))) float v2f;
typedef __attribute__((ext_vector_type(8))) float v8f;

#define N_PATCH 196
#define FEAT    784   // 196*4
#define FROW    788   // LDS row stride: 788%64=20 -> 16 distinct banks over rows,
                      // and 788*4 bytes keeps rows 16B-aligned for float4 stores

// lane i <- lane (i ^ 16): v_permlanex16_b32 with identity lane selects
__device__ __forceinline__ float xor16(float v) {
    int r = __builtin_amdgcn_permlanex16(
        0, __float_as_int(v), 0x76543210u, 0xfedcba98u, false, false);
    return __int_as_float(r);
}

__device__ __forceinline__ float half_sum(float v) {
    v += __shfl_xor(v, 1);
    v += __shfl_xor(v, 2);
    v += __shfl_xor(v, 4);
    v += __shfl_xor(v, 8);
    return v;
}

__device__ __forceinline__ float half_max(float v) {
    v = fmaxf(v, __shfl_xor(v, 1));
    v = fmaxf(v, __shfl_xor(v, 2));
    v = fmaxf(v, __shfl_xor(v, 4));
    v = fmaxf(v, __shfl_xor(v, 8));
    return v;
}

__global__ __launch_bounds__(256)
void quanv_fused_kernel(const float* __restrict__ x,   // (B,1,28,28)
                        const float* __restrict__ U,   // (16,16)
                        const float* __restrict__ W,   // (10,784)
                        const float* __restrict__ bias,// (10,)
                        float* __restrict__ out) {     // (B,10) log_softmax
    __shared__ float featLds[16 * FROW];  // 16 images x 784 features (padded)
    __shared__ float accLds[16 * 16];     // logits partial accumulator [M][N]

    const int tid  = threadIdx.x;
    const int wave = tid >> 5;
    const int lane = tid & 31;
    const int n16  = lane & 15;           // image-in-block / N index
    const int hi   = lane >> 4;           // half-wave select
    const int img0 = blockIdx.x * 16;     // first image of this block

    accLds[tid] = 0.0f;                   // zeroed before phase-B shared atomics

    const float sgnH = hi ? -1.0f : 1.0f; // (-1)^bit3(e) for qubit-0 measurement

    // ---------------- Phase A: Z = U x S^T via WMMA f32 16x16x4 ----------------
    // A = U: lane = row m = e_out = n16, K = 4*c4 + 2*hi + v  (loop-invariant)
    float ua[8];
#pragma unroll
    for (int c4 = 0; c4 < 4; ++c4) {
        ua[2 * c4 + 0] = U[n16 * 16 + 4 * c4 + 2 * hi + 0];
        ua[2 * c4 + 1] = U[n16 * 16 + 4 * c4 + 2 * hi + 1];
    }

    // B = S^T: lane = image col n16 -> this lane builds that image's state
    const float* Ximg = x + (size_t)(img0 + n16) * 784;

    for (int p = wave; p < N_PATCH; p += 8) {   // wave-uniform loop
        const int r   = p / 14;
        const int cc  = p - 14 * r;
        const int off = (2 * r) * 28 + 2 * cc;
        // patch element order: [x(r,c), x(r,c+1), x(r+1,c), x(r+1,c+1)]
        float th0 = 0.5f * Ximg[off];
        float th1 = 0.5f * Ximg[off + 1];
        float th2 = 0.5f * Ximg[off + 28];
        float th3 = 0.5f * Ximg[off + 29];
        float c0, s0, c1, s1, c2, s2, c3, s3;
        __sincosf(th0, &s0, &c0);
        __sincosf(th1, &s1, &c1);
        __sincosf(th2, &s2, &c2);
        __sincosf(th3, &s3, &c3);

        // state amp st[K] = (K&8?s0:c0)(K&4?s1:c1)(K&2?s2:c2)(K&1?s3:c3);
        // K = 4*c4 + 2*hi + v: bits3..2 = c4 (compile-time), bit1 = hi, bit0 = v.
        const float f2 = hi ? s2 : c2;      // single runtime select, hoisted
        const float g0 = f2 * c3;           // v = 0
        const float g1 = f2 * s3;           // v = 1
        const float p01[4] = {c0 * c1, c0 * s1, s0 * c1, s0 * s1}; // by c4

        v8f acc = {0.f, 0.f, 0.f, 0.f, 0.f, 0.f, 0.f, 0.f};
#pragma unroll
        for (int c4 = 0; c4 < 4; ++c4) {
            v2f a;                          // A = U tile for this K chunk
            a.x = ua[2 * c4 + 0];
            a.y = ua[2 * c4 + 1];
            v2f bm;                         // B = S^T: col n16, K = 4c4+2hi+v
            bm.x = p01[c4] * g0;
            bm.y = p01[c4] * g1;
            acc = __builtin_amdgcn_wmma_f32_16x16x4_f32(
                false, a, false, bm, (short)0, acc, false, false);
        }

        // D: lane = image n16, slot v -> e = v + 8*hi.  probs p_v = acc[v]^2.
        float p0 = acc[0] * acc[0], p1 = acc[1] * acc[1];
        float p2 = acc[2] * acc[2], p3 = acc[3] * acc[3];
        float p4 = acc[4] * acc[4], p5 = acc[5] * acc[5];
        float p6 = acc[6] * acc[6], p7 = acc[7] * acc[7];

        // compile-time-signed adder tree (e bits 2..0 == v bits 2..0)
        float s01 = p0 + p1, d01 = p0 - p1;
        float s23 = p2 + p3, d23 = p2 - p3;
        float s45 = p4 + p5, d45 = p4 - p5;
        float s67 = p6 + p7, d67 = p6 - p7;
        float sA  = s01 + s23, sB = s45 + s67;
        float part0 = sA + sB;                         // plain sum (qubit 0 uses half sign)
        float part1 = sA - sB;                         // sign by e bit2 (qubit 1)
        float part2 = (s01 - s23) + (s45 - s67);       // sign by e bit1 (qubit 2)
        float part3 = (d01 + d23) + (d45 + d67);       // sign by e bit0 (qubit 3)

        // combine the two halves (e bit3 = hi): VALU lane-xor-16 swaps
        float t0 = xor16(part0);
        float t1 = xor16(part1);
        float t2 = xor16(part2);
        float t3 = xor16(part3);
        float m0 = sgnH * (part0 - t0);  // qubit 0: lo-half minus hi-half
        float m1 = part1 + t1;           // qubit 1..3: same sign both halves
        float m2 = part2 + t2;
        float m3 = part3 + t3;

        if (hi == 0) {                   // one vectorized store per image
            float4 mv = make_float4(m0, m1, m2, m3);
            *(float4*)&featLds[n16 * FROW + 4 * p] = mv;
        }
    }

    __syncthreads();

    // ---------------- Phase B: logits GEMM, K split across waves ----------------
    v8f lacc = {0.f, 0.f, 0.f, 0.f, 0.f, 0.f, 0.f, 0.f};
    const bool nvalid = (n16 < 10);
#pragma unroll 2
    for (int c = wave; c < FEAT / 4; c += 8) {  // 196 K-chunks, wave-uniform
        const int k = 4 * c + 2 * hi;
        v2f a;                                   // A: feat row M=n16 (ds_load_b64)
        a.x = featLds[n16 * FROW + k];
        a.y = featLds[n16 * FROW + k + 1];
        v2f bm;                                  // B: W^T, col n=n16 (0 for n>=10)
        bm.x = nvalid ? W[n16 * FEAT + k]     : 0.0f;
        bm.y = nvalid ? W[n16 * FEAT + k + 1] : 0.0f;
        lacc = __builtin_amdgcn_wmma_f32_16x16x4_f32(
            false, a, false, bm, (short)0, lacc, false, false);
    }
#pragma unroll
    for (int v = 0; v < 8; ++v)
        atomicAdd(&accLds[(v + 8 * hi) * 16 + n16], lacc[v]);  // ds_add_f32

    __syncthreads();

    // ---------------- bias + log_softmax (wave 0) ----------------
    if (tid < 32) {
#pragma unroll
        for (int v = 0; v < 8; ++v) {
            const int row = v + 8 * hi;          // image within block
            float logit = accLds[row * 16 + n16] + (nvalid ? bias[n16] : 0.0f);
            float mx = half_max(nvalid ? logit : -INFINITY);
            float e  = nvalid ? __expf(logit - mx) : 0.0f;
            float s  = half_sum(e);
            float res = logit - mx - __logf(s);
            if (nvalid)
                out[(size_t)(img0 + row) * 10 + n16] = res;
        }
    }
}

extern "C" void kernel_launch(void* const* d_in, const int* in_sizes, int n_in,
                              void* d_out, int out_size, void* d_ws, size_t ws_size,
                              hipStream_t stream) {
    (void)n_in; (void)d_ws; (void)ws_size; (void)out_size;
    const float* x    = (const float*)d_in[0];  // (B,1,28,28)
    const float* U    = (const float*)d_in[1];  // (16,16)
    const float* W    = (const float*)d_in[2];  // (10,784)
    const float* bias = (const float*)d_in[3];  // (10,)
    float* out = (float*)d_out;

    const int B = in_sizes[0] / 784;            // 8192
    const int blocks = B / 16;                  // 512 (16 images per block)
    quanv_fused_kernel<<<blocks, 256, 0, stream>>>(x, U, W, bias, out);
}